// FCOS_22393959481808
// MI455X (gfx1250) — compile-verified
//
#include <hip/hip_runtime.h>
#include <hip/hip_bf16.h>
#include <math.h>

// ---------------------------------------------------------------------------
// FCOS box decode + class-offset NMS for gfx1250 (MI455X).
//   K1: decode boxes + per-block coordinate max
//   K2: final max reduction -> max_coordinate
//   K3: O(N^2) stable rank (argsort -scores) + scatter sorted NMS arrays
//       (async global->LDS staging of score tiles, ASYNCcnt-tracked)
//   K4: pairwise suppression bitmask via V_WMMA_F32_16X16X4_F32 outer ops
//   K5: serial greedy scan (keep bits in LDS) + output scatter, with
//       global_prefetch of the next mask row
// ---------------------------------------------------------------------------

typedef __attribute__((ext_vector_type(2))) float v2f;
typedef __attribute__((ext_vector_type(8))) float v8f;

#if defined(__AMDGCN__) && __has_builtin(__builtin_amdgcn_wmma_f32_16x16x4_f32)
#define HAVE_WMMA4 1
#else
#define HAVE_WMMA4 0
#endif

#if defined(__AMDGCN__) && __has_builtin(__builtin_amdgcn_global_load_async_to_lds_b32) && __has_builtin(__builtin_amdgcn_s_wait_asynccnt)
#define HAVE_ASYNC_LDS 1
#else
#define HAVE_ASYNC_LDS 0
#endif

__device__ __forceinline__ unsigned ballot32(bool p) {
#if defined(__AMDGCN__) && __has_builtin(__builtin_amdgcn_ballot_w32)
  return __builtin_amdgcn_ballot_w32(p);
#else
  return p ? 1u : 0u;  // host-pass stub
#endif
}

#if HAVE_WMMA4
// D[i][j] = a.x_i * b.x_j + a.y_i * b.y_j  (16x16 f32 tile, K=4 with K2/K3 = 0)
__device__ __forceinline__ v8f outer4(v2f a, v2f b) {
  v8f c = {0.f, 0.f, 0.f, 0.f, 0.f, 0.f, 0.f, 0.f};
  return __builtin_amdgcn_wmma_f32_16x16x4_f32(false, a, false, b, (short)0, c,
                                               false, false);
}
#endif

// ---------------------------------------------------------------- K1: decode
__global__ void k_boxes(const float* __restrict__ deltas,
                        const float* __restrict__ locs,
                        const int* __restrict__ stridep,
                        float* __restrict__ boxes_out,
                        float* __restrict__ blockmax, int N) {
  __shared__ float red[256];
  int i = blockIdx.x * 256 + threadIdx.x;
  float m = -3.0e38f;
  if (i < N) {
    float s  = (float)stridep[0];
    float d0 = fmaxf(deltas[i * 4 + 0], 0.f);
    float d1 = fmaxf(deltas[i * 4 + 1], 0.f);
    float d2 = fmaxf(deltas[i * 4 + 2], 0.f);
    float d3 = fmaxf(deltas[i * 4 + 3], 0.f);
    float xc = locs[i * 2 + 0], yc = locs[i * 2 + 1];
    float x1 = xc - s * d0, y1 = yc - s * d1;
    float x2 = xc + s * d2, y2 = yc + s * d3;
    boxes_out[i * 4 + 0] = x1;
    boxes_out[i * 4 + 1] = y1;
    boxes_out[i * 4 + 2] = x2;
    boxes_out[i * 4 + 3] = y2;
    m = fmaxf(fmaxf(x1, y1), fmaxf(x2, y2));
  }
  red[threadIdx.x] = m;
  __syncthreads();
  for (int st = 128; st > 0; st >>= 1) {
    if (threadIdx.x < st)
      red[threadIdx.x] = fmaxf(red[threadIdx.x], red[threadIdx.x + st]);
    __syncthreads();
  }
  if (threadIdx.x == 0) blockmax[blockIdx.x] = red[0];
}

// ------------------------------------------------------------- K2: final max
__global__ void k_rmax(const float* __restrict__ bm, float* __restrict__ maxc,
                       int nb) {
  __shared__ float red[64];
  int t = threadIdx.x;
  float m = -3.0e38f;
  for (int k = t; k < nb; k += 64) m = fmaxf(m, bm[k]);
  red[t] = m;
  __syncthreads();
  for (int s = 32; s > 0; s >>= 1) {
    if (t < s) red[t] = fmaxf(red[t], red[t + s]);
    __syncthreads();
  }
  if (t == 0) maxc[0] = red[0];
}

// --------------------------------------------- K3: stable rank + sorted scatter
__global__ void k_rank(const float* __restrict__ scores,
                       const int* __restrict__ cls,
                       const float* __restrict__ boxes,
                       const float* __restrict__ maxcp,
                       float* __restrict__ X1, float* __restrict__ Y1,
                       float* __restrict__ X2, float* __restrict__ Y2,
                       float* __restrict__ Wd, float* __restrict__ Hd,
                       float* __restrict__ Ar, unsigned* __restrict__ order,
                       int N) {
  __shared__ float st[256];
  int t = threadIdx.x;
  int i = blockIdx.x * 256 + t;
  float si = (i < N) ? scores[i] : 0.f;
  int rank = 0;
  for (int jt = 0; jt < N; jt += 256) {
#if HAVE_ASYNC_LDS
    {
      typedef __attribute__((address_space(1))) int gint;
      typedef __attribute__((address_space(3))) int lint;
      __builtin_amdgcn_global_load_async_to_lds_b32(
          (gint*)(scores + jt + t), (lint*)(&st[t]), 0, 0);
      __builtin_amdgcn_s_wait_asynccnt(0);
    }
    __syncthreads();
#else
    st[t] = scores[jt + t];
    __syncthreads();
#endif
#pragma unroll 8
    for (int k = 0; k < 256; ++k) {
      float sj = st[k];
      int j = jt + k;
      rank += ((sj > si) || ((sj == si) && (j < i))) ? 1 : 0;
    }
    __syncthreads();
  }
  if (i < N) {
    float mc = maxcp[0];
    float off = (float)cls[i] * (mc + 1.0f);
    float x1 = boxes[i * 4 + 0], y1 = boxes[i * 4 + 1];
    float x2 = boxes[i * 4 + 2], y2 = boxes[i * 4 + 3];
    float w = x2 - x1, h = y2 - y1;
    order[rank] = (unsigned)i;
    X1[rank] = x1 + off;
    Y1[rank] = y1 + off;
    X2[rank] = x2 + off;
    Y2[rank] = y2 + off;
    Wd[rank] = w;
    Hd[rank] = h;
    Ar[rank] = w * h;
  }
}

// ------------------------------------------ K4: suppression bitmask via WMMA
// grid: (N/16, N/256), block 256 (8 waves). Wave handles 16 rows x 32 cols
// (= one u32 mask word per row). Bit set <=> iou>0.5 && j>i.
__global__ void k_mask(const float* __restrict__ X1, const float* __restrict__ Y1,
                       const float* __restrict__ X2, const float* __restrict__ Y2,
                       const float* __restrict__ Wd, const float* __restrict__ Hd,
                       const float* __restrict__ Ar, unsigned* __restrict__ mask,
                       int N) {
  const int nw = N >> 5;
  const int lane = threadIdx.x & 31;
  const int wave = threadIdx.x >> 5;
  const int i0 = blockIdx.x << 4;
  const int jp = blockIdx.y * 8 + wave;
  const int j0 = jp << 5;

  if (j0 + 31 <= i0) {  // entire 32-col word is j<=i for every row: all zero
    if (lane < 16) mask[(size_t)(i0 + lane) * nw + jp] = 0u;
    return;
  }

  const bool lo = lane < 16;
  const int li = lane & 15;
  unsigned bits0[8], bits1[8];

#if HAVE_WMMA4
  // A fragment (16x4 f32): lanes 0-15 carry K0/K1 rows, lanes 16-31 = K2/K3 = 0
  const int ri = i0 + li;
  const float kneg = lo ? -1.f : 0.f;
  const float kpos = lo ? 1.f : 0.f;
  v2f aX1 = {lo ? X1[ri] : 0.f, kneg};
  v2f aX2 = {lo ? X2[ri] : 0.f, kneg};
  v2f aY1 = {lo ? Y1[ri] : 0.f, kneg};
  v2f aY2 = {lo ? Y2[ri] : 0.f, kneg};
  v2f aW  = {lo ? Wd[ri] : 0.f, kpos};
  v2f aH  = {lo ? Hd[ri] : 0.f, kpos};
  v2f aA  = {lo ? Ar[ri] : 0.f, kpos};

#pragma unroll
  for (int tt = 0; tt < 2; ++tt) {
    const int cj = j0 + tt * 16 + li;
    v2f bX1 = {kpos, lo ? X1[cj] : 0.f};
    v2f bX2 = {kpos, lo ? X2[cj] : 0.f};
    v2f bY1 = {kpos, lo ? Y1[cj] : 0.f};
    v2f bY2 = {kpos, lo ? Y2[cj] : 0.f};
    v2f bW  = {kpos, lo ? Wd[cj] : 0.f};
    v2f bH  = {kpos, lo ? Hd[cj] : 0.f};
    v2f bA  = {kpos, lo ? Ar[cj] : 0.f};

    v8f dx1 = outer4(aX1, bX1);  // x1_i - x1_j
    v8f dx2 = outer4(aX2, bX2);  // x2_i - x2_j
    v8f sx  = outer4(aW, bW);    // w_i + w_j
    v8f dy1 = outer4(aY1, bY1);
    v8f dy2 = outer4(aY2, bY2);
    v8f sy  = outer4(aH, bH);
    v8f sa  = outer4(aA, bA);    // area_i + area_j

    unsigned* bp = tt ? bits1 : bits0;
#pragma unroll
    for (int r = 0; r < 8; ++r) {
      // 2*w = (w_i+w_j) - |x2_i-x2_j| - |x1_i-x1_j|, clamped
      float w2 = fmaxf(sx[r] - fabsf(dx1[r]) - fabsf(dx2[r]), 0.f);
      float h2 = fmaxf(sy[r] - fabsf(dy1[r]) - fabsf(dy2[r]), 0.f);
      // iou>0.5 <=> 3*inter > area_i+area_j <=> 0.75*(2w)(2h) > sa
      bool p = (0.75f * w2 * h2) > sa[r];
      bp[r] = ballot32(p);
    }
  }
#else
  for (int tt = 0; tt < 2; ++tt) {
    const int jj = j0 + tt * 16 + li;
    float bx1 = X1[jj], bx2 = X2[jj], by1 = Y1[jj], by2 = Y2[jj], ba = Ar[jj];
    unsigned* bp = tt ? bits1 : bits0;
#pragma unroll
    for (int r = 0; r < 8; ++r) {
      int ii = i0 + r + (lo ? 0 : 8);
      float w = fmaxf(fminf(X2[ii], bx2) - fmaxf(X1[ii], bx1), 0.f);
      float h = fmaxf(fminf(Y2[ii], by2) - fmaxf(Y1[ii], by1), 0.f);
      bool p = 3.f * w * h > (Ar[ii] + ba);
      bp[r] = ballot32(p);
    }
  }
#endif

  // ballot bits [0:16) = row r, cols jb..jb+15 ; bits [16:32) = row r+8
  if (lane == 0) {
#pragma unroll
    for (int r = 0; r < 8; ++r) {
      int row0 = i0 + r, row1 = i0 + r + 8;
      unsigned w0 = (bits0[r] & 0xFFFFu) | (bits1[r] << 16);
      unsigned w1 = (bits0[r] >> 16) | (bits1[r] & 0xFFFF0000u);
      unsigned a0 = 0xFFFFFFFFu, a1 = 0xFFFFFFFFu;  // enforce j > i
      if (row0 >= j0) {
        int sh = row0 - j0 + 1;
        a0 = (sh >= 32) ? 0u : (0xFFFFFFFFu << sh);
      }
      if (row1 >= j0) {
        int sh = row1 - j0 + 1;
        a1 = (sh >= 32) ? 0u : (0xFFFFFFFFu << sh);
      }
      mask[(size_t)row0 * nw + jp] = w0 & a0;
      mask[(size_t)row1 * nw + jp] = w1 & a1;
    }
  }
}

// --------------------------------------------- K5: serial greedy scan + output
__global__ void k_scan(const unsigned* __restrict__ mask,
                       const unsigned* __restrict__ order,
                       float* __restrict__ keep_out,
                       float* __restrict__ order_out, int N) {
  __shared__ unsigned keep[256];
  __shared__ int cur;
  const int t = threadIdx.x;
  const int nw = N >> 5;
  keep[t] = 0xFFFFFFFFu;
  __syncthreads();
  for (int i = 0; i < N; ++i) {
    if (t == 0) cur = (keep[i >> 5] >> (i & 31)) & 1;
    __syncthreads();
    if (i + 1 < N)
      __builtin_prefetch(&mask[(size_t)(i + 1) * nw + t], 0, 1);
    if (cur) keep[t] &= ~mask[(size_t)i * nw + t];
    __syncthreads();
  }
  for (int k = t; k < N; k += 256) {
    unsigned bit = (keep[k >> 5] >> (k & 31)) & 1u;
    unsigned oi = order[k];
    keep_out[oi] = (float)bit;   // keep_orig scattered back to input order
    order_out[k] = (float)oi;    // argsort(-scores)
  }
}

// ---------------------------------------------------------------------------
extern "C" void kernel_launch(void* const* d_in, const int* in_sizes, int n_in,
                              void* d_out, int out_size, void* d_ws,
                              size_t ws_size, hipStream_t stream) {
  const float* deltas = (const float*)d_in[0];
  const float* locs   = (const float*)d_in[1];
  const float* scores = (const float*)d_in[2];
  const int*   cls    = (const int*)d_in[3];
  const int*   stridep = (const int*)d_in[4];

  const int N = in_sizes[2];  // 8192

  float* out       = (float*)d_out;
  float* outBoxes  = out;            // N*4
  float* outKeep   = out + 4 * N;    // N
  float* outOrder  = out + 5 * N;    // N

  // workspace layout (floats/u32 words)
  float* wsf      = (float*)d_ws;
  float* maxc     = wsf;             // [1]
  float* blockmax = wsf + 32;        // [32]
  float* X1 = wsf + 64;
  float* Y1 = X1 + N;
  float* X2 = Y1 + N;
  float* Y2 = X2 + N;
  float* Wd = Y2 + N;
  float* Hd = Wd + N;
  float* Ar = Hd + N;
  unsigned* order = (unsigned*)(Ar + N);
  unsigned* mask  = order + N;       // N * (N/32) words = 8 MB for N=8192

  const int nb = (N + 255) / 256;
  k_boxes<<<nb, 256, 0, stream>>>(deltas, locs, stridep, outBoxes, blockmax, N);
  k_rmax<<<1, 64, 0, stream>>>(blockmax, maxc, nb);
  k_rank<<<nb, 256, 0, stream>>>(scores, cls, outBoxes, maxc, X1, Y1, X2, Y2,
                                 Wd, Hd, Ar, order, N);
  dim3 gmask(N / 16, N / 256);
  k_mask<<<gmask, 256, 0, stream>>>(X1, Y1, X2, Y2, Wd, Hd, Ar, mask, N);
  k_scan<<<1, 256, 0, stream>>>(mask, order, outKeep, outOrder, N);
}